// StrictLowerTriQSM_1374389534764
// MI455X (gfx1250) — compile-verified
//
#include <hip/hip_runtime.h>

// StrictLowerTriQSM via chunked associative scan on gfx1250.
// State f (8x16, f32) lives in WMMA B-operand layout (4 floats per lane of a
// wave32). All 8x8 * 8x16 products use V_WMMA_F32_16X16X4_F32 (K split 4+4),
// rank-1 update outer(q,x) is folded in as a third K=4 WMMA.

typedef __attribute__((ext_vector_type(2))) float v2f;
typedef __attribute__((ext_vector_type(8))) float v8f;

#define CHUNK 256   // elements per chunk (one wave each)
#define GSIZE 64    // chunks per group in level-2 scan

static __device__ __forceinline__ v8f wmma4(v2f a, v2f b, v8f c) {
  // D = A(16x4) * B(4x16) + C(16x16), all f32
  return __builtin_amdgcn_wmma_f32_16x16x4_f32(false, a, false, b, (short)0, c,
                                               false, false);
}

// State element mapping: lane L holds tile[k][n] for i=0..3 where
//   n = L & 15, h = L >> 4, k = (i>>1)*4 + h*2 + (i&1)
static __device__ __forceinline__ int kOf(int i, int h) {
  return ((i >> 1) << 2) + (h << 1) + (i & 1);
}

// Convert WMMA D (16x16, rows 8..15 == 0) back into the 4-float state.
static __device__ __forceinline__ void dToState(v8f d, float s[4], int lane) {
  int src = lane & 15;
  float t0 = __shfl(d[2], src);
  float t1 = __shfl(d[3], src);
  float t2 = __shfl(d[6], src);
  float t3 = __shfl(d[7], src);
  bool lo = lane < 16;
  s[0] = lo ? d[0] : t0;
  s[1] = lo ? d[1] : t1;
  s[2] = lo ? d[4] : t2;
  s[3] = lo ? d[5] : t3;
}

// Build the two 16x4 A-operands (cols 0..3 and 4..7) of an 8x8 row-major
// matrix, with rows 8..15 zeroed.
static __device__ __forceinline__ void loadAops(const float* __restrict__ a,
                                                int lane, v2f& A0, v2f& A1) {
  int m = lane & 15;
  int c0 = (lane >> 4) << 1;  // 0 or 2
  const float* r = a + (m & 7) * 8 + c0;
  float a00 = r[0], a01 = r[1], a10 = r[4], a11 = r[5];
  bool v = m < 8;
  A0[0] = v ? a00 : 0.f;  A0[1] = v ? a01 : 0.f;
  A1[0] = v ? a10 : 0.f;  A1[1] = v ? a11 : 0.f;
}

// s <- A * s + accInit (accInit is a D-layout tile, e.g. rank-1 term or zero)
static __device__ __forceinline__ void applyA(v2f A0, v2f A1, float s[4],
                                              int lane, v8f acc) {
  v2f B0; B0[0] = s[0]; B0[1] = s[1];
  v2f B1; B1[0] = s[2]; B1[1] = s[3];
  acc = wmma4(A0, B0, acc);
  acc = wmma4(A1, B1, acc);
  dToState(acc, s, lane);
}

// ---------------- pass 1: per-chunk carries (A_c, B_c) ----------------
__global__ __launch_bounds__(32) void qsm_pass1(
    const float* __restrict__ q, const float* __restrict__ a,
    const float* __restrict__ x, float* __restrict__ wsA,
    float* __restrict__ wsB, int n, int nChunks) {
  int c = blockIdx.x;
  if (c >= nChunks) return;
  int lane = threadIdx.x & 31;
  int h = lane >> 4, n0 = lane & 15;
  int start = c * CHUNK;
  int cnt = min(CHUNK, n - start);
  float sF[4] = {0.f, 0.f, 0.f, 0.f};
  float sP[4];
#pragma unroll
  for (int i = 0; i < 4; ++i) sP[i] = (kOf(i, h) == n0) ? 1.f : 0.f;  // I
  const float* ap = a + (size_t)start * 64;
  const float* qp = q + (size_t)start * 8;
  const float* xp = x + (size_t)start * 16;
  for (int t = 0; t < cnt; ++t) {
    v2f A0, A1;
    loadAops(ap + (size_t)t * 64, lane, A0, A1);
    float qv = qp[t * 8 + (lane & 7)];
    float xv = xp[t * 16 + n0];
    v2f Aq; Aq[0] = (lane < 8) ? qv : 0.f;  Aq[1] = 0.f;
    v2f Bx; Bx[0] = (lane < 16) ? xv : 0.f; Bx[1] = 0.f;
    v8f z = {};
    v8f r1 = wmma4(Aq, Bx, z);      // outer(q,x)
    applyA(A0, A1, sF, lane, r1);   // f = a*f + q x^T
    applyA(A0, A1, sP, lane, z);    // P = a*P
  }
#pragma unroll
  for (int i = 0; i < 4; ++i) {
    int k = kOf(i, h);
    wsB[(size_t)c * 128 + k * 16 + n0] = sF[i];
    if (n0 < 8) wsA[(size_t)c * 64 + k * 8 + n0] = sP[i];
  }
}

// ------- pass 2a: within-group exclusive prefixes + group totals -------
__global__ __launch_bounds__(32) void qsm_pass2a(
    const float* __restrict__ wsA, const float* __restrict__ wsB,
    float* __restrict__ wsEA, float* __restrict__ wsEB,
    float* __restrict__ wsGA, float* __restrict__ wsGB, int nChunks,
    int nGroups) {
  int g = blockIdx.x;
  if (g >= nGroups) return;
  int lane = threadIdx.x & 31;
  int h = lane >> 4, n0 = lane & 15;
  int cs = g * GSIZE, ce = min(cs + GSIZE, nChunks);
  float sA[4], sB[4] = {0.f, 0.f, 0.f, 0.f};
#pragma unroll
  for (int i = 0; i < 4; ++i) sA[i] = (kOf(i, h) == n0) ? 1.f : 0.f;
  for (int c = cs; c < ce; ++c) {
    // store exclusive prefix (state BEFORE composing chunk c)
#pragma unroll
    for (int i = 0; i < 4; ++i) {
      int k = kOf(i, h);
      wsEB[(size_t)c * 128 + k * 16 + n0] = sB[i];
      if (n0 < 8) wsEA[(size_t)c * 64 + k * 8 + n0] = sA[i];
    }
    v2f A0, A1;
    loadAops(wsA + (size_t)c * 64, lane, A0, A1);
    v8f z = {};
    applyA(A0, A1, sB, lane, z);  // B' = A_c * B
#pragma unroll
    for (int i = 0; i < 4; ++i)
      sB[i] += wsB[(size_t)c * 128 + kOf(i, h) * 16 + n0];  // + B_c
    applyA(A0, A1, sA, lane, z);  // A' = A_c * A
  }
#pragma unroll
  for (int i = 0; i < 4; ++i) {
    int k = kOf(i, h);
    wsGB[(size_t)g * 128 + k * 16 + n0] = sB[i];
    if (n0 < 8) wsGA[(size_t)g * 64 + k * 8 + n0] = sA[i];
  }
}

// ---------- pass 2b: scan over group totals -> group start states ----------
__global__ __launch_bounds__(32) void qsm_pass2b(
    const float* __restrict__ wsGA, const float* __restrict__ wsGB,
    float* __restrict__ wsGS, int nGroups) {
  if (blockIdx.x != 0) return;
  int lane = threadIdx.x & 31;
  int h = lane >> 4, n0 = lane & 15;
  float sF[4] = {0.f, 0.f, 0.f, 0.f};
  for (int g = 0; g < nGroups; ++g) {
#pragma unroll
    for (int i = 0; i < 4; ++i)
      wsGS[(size_t)g * 128 + kOf(i, h) * 16 + n0] = sF[i];
    v2f A0, A1;
    loadAops(wsGA + (size_t)g * 64, lane, A0, A1);
    v8f z = {};
    applyA(A0, A1, sF, lane, z);
#pragma unroll
    for (int i = 0; i < 4; ++i)
      sF[i] += wsGB[(size_t)g * 128 + kOf(i, h) * 16 + n0];
  }
}

// ------------- pass 3: replay chunks with exact f_in, emit y -------------
__global__ __launch_bounds__(32) void qsm_pass3(
    const float* __restrict__ p, const float* __restrict__ q,
    const float* __restrict__ a, const float* __restrict__ x,
    const float* __restrict__ wsEA, const float* __restrict__ wsEB,
    const float* __restrict__ wsGS, float* __restrict__ out, int n,
    int nChunks) {
  int c = blockIdx.x;
  if (c >= nChunks) return;
  int g = c / GSIZE;
  int lane = threadIdx.x & 31;
  int h = lane >> 4, n0 = lane & 15;
  float sF[4];
#pragma unroll
  for (int i = 0; i < 4; ++i)
    sF[i] = wsGS[(size_t)g * 128 + kOf(i, h) * 16 + n0];
  {  // f_in = E_A(c) * f_groupstart + E_B(c)
    v2f A0, A1;
    loadAops(wsEA + (size_t)c * 64, lane, A0, A1);
    v8f z = {};
    applyA(A0, A1, sF, lane, z);
#pragma unroll
    for (int i = 0; i < 4; ++i)
      sF[i] += wsEB[(size_t)c * 128 + kOf(i, h) * 16 + n0];
  }
  int start = c * CHUNK;
  int cnt = min(CHUNK, n - start);
  const float* ap = a + (size_t)start * 64;
  const float* qp = q + (size_t)start * 8;
  const float* xp = x + (size_t)start * 16;
  const float* pp = p + (size_t)start * 8;
  float* op = out + (size_t)start * 16;
  for (int t = 0; t < cnt; ++t) {
    // y_t = p_t^T f  (f BEFORE this step's update)
    float pv = pp[t * 8 + (lane & 7)];
    float partial = 0.f;
#pragma unroll
    for (int i = 0; i < 4; ++i) partial += __shfl(pv, kOf(i, h)) * sF[i];
    float other = __shfl(partial, n0 + 16);
    if (lane < 16) op[t * 16 + lane] = partial + other;
    // f = a_t * f + q_t x_t^T
    v2f A0, A1;
    loadAops(ap + (size_t)t * 64, lane, A0, A1);
    float qv = qp[t * 8 + (lane & 7)];
    float xv = xp[t * 16 + n0];
    v2f Aq; Aq[0] = (lane < 8) ? qv : 0.f;  Aq[1] = 0.f;
    v2f Bx; Bx[0] = (lane < 16) ? xv : 0.f; Bx[1] = 0.f;
    v8f z = {};
    v8f r1 = wmma4(Aq, Bx, z);
    applyA(A0, A1, sF, lane, r1);
  }
}

extern "C" void kernel_launch(void* const* d_in, const int* in_sizes, int n_in,
                              void* d_out, int out_size, void* d_ws,
                              size_t ws_size, hipStream_t stream) {
  (void)n_in; (void)out_size; (void)ws_size;
  const float* p = (const float*)d_in[0];  // (N,8)
  const float* q = (const float*)d_in[1];  // (N,8)
  const float* a = (const float*)d_in[2];  // (N,8,8)
  const float* x = (const float*)d_in[3];  // (N,16)
  int n = in_sizes[0] / 8;
  int nChunks = (n + CHUNK - 1) / CHUNK;
  int nGroups = (nChunks + GSIZE - 1) / GSIZE;

  float* wsA  = (float*)d_ws;                       // carry A: C*64
  float* wsB  = wsA  + (size_t)nChunks * 64;        // carry B: C*128
  float* wsEA = wsB  + (size_t)nChunks * 128;       // excl prefix A: C*64
  float* wsEB = wsEA + (size_t)nChunks * 64;        // excl prefix B: C*128
  float* wsGA = wsEB + (size_t)nChunks * 128;       // group total A: G*64
  float* wsGB = wsGA + (size_t)nGroups * 64;        // group total B: G*128
  float* wsGS = wsGB + (size_t)nGroups * 128;       // group start f: G*128

  qsm_pass1<<<nChunks, 32, 0, stream>>>(q, a, x, wsA, wsB, n, nChunks);
  qsm_pass2a<<<nGroups, 32, 0, stream>>>(wsA, wsB, wsEA, wsEB, wsGA, wsGB,
                                         nChunks, nGroups);
  qsm_pass2b<<<1, 32, 0, stream>>>(wsGA, wsGB, wsGS, nGroups);
  qsm_pass3<<<nChunks, 32, 0, stream>>>(p, q, a, x, wsEA, wsEB, wsGS,
                                        (float*)d_out, n, nChunks);
}